// RNN_69767448756331
// MI455X (gfx1250) — compile-verified
//
#include <hip/hip_runtime.h>
#include <math.h>

// Problem sizes (fixed by the reference)
#define NB 64      // batch
#define TT 512     // timesteps
#define DD 512     // input dim
#define HH 512     // hidden dim

#define KC   16    // K-chunk staged in LDS per iteration
#define LPAD 20    // padded LDS row stride in words: (20*lane) mod 64 distinct
                   // for 16 lanes -> conflict-free b64 fragment reads; even ->
                   // 8B alignment preserved.

typedef float v2f __attribute__((ext_vector_type(2)));
typedef float v4f __attribute__((ext_vector_type(4)));
typedef float v8f __attribute__((ext_vector_type(8)));

// One fp32 WMMA k-step: C += A(16x4) * B(4x16)
// A frag (ISA 7.12.2): lane lr<16 rows M=lr, VGPR{0,1}=K{0,1}; lanes 16-31 K{2,3}
//   -> per-lane float2 from LDS-A at [mw+lr][kk + 2*half]
// B frag: VGPR0 = row K=2*half col n0+lr, VGPR1 = row K=2*half+1
//   -> stored TRANSPOSED in LDS (BT[n][k]) so it is one contiguous float2.
// C/D: VGPR e -> row m0 + e + 8*half, col n0 + lr.
#define WMMA_STEP(cacc, bcol)                                                      \
    do {                                                                           \
        v2f b = *(const v2f*)(&lBT[(size_t)((bcol) + lr) * LPAD + kk + 2 * half]); \
        cacc = __builtin_amdgcn_wmma_f32_16x16x4_f32(false, a, false, b,           \
                                                     (short)0, cacc, false, false);\
    } while (0)

// ---------------------------------------------------------------------------
// Kernel 1: XW[m, h] = sum_d X[m, d] * Wx[d, h] + bias[h],  M = NB*TT
// Block: 256 threads = 8 waves. Block tile 128(M) x 64(N); wave w owns rows
// w*16..w*16+15 and all 64 cols (4 accumulators). B fragments are shared by
// all 8 waves through LDS. Grid: (HH/64, M/128).
// ---------------------------------------------------------------------------
__global__ __launch_bounds__(256) void xw_gemm_kernel(
    const float* __restrict__ X,     // (M, DD)
    const float* __restrict__ Wx,    // (DD, HH)
    const float* __restrict__ bias,  // (HH)
    float* __restrict__ XW)          // (M, HH)
{
    __shared__ float lA[128 * LPAD];  // A tile: [row][k], padded
    __shared__ float lBT[64 * LPAD];  // B tile transposed: [col][k], padded

    const int tid  = threadIdx.x;
    const int wave = tid >> 5;
    const int lane = tid & 31;
    const int half = lane >> 4;
    const int lr   = lane & 15;

    const int n0 = blockIdx.x * 64;
    const int m0 = blockIdx.y * 128;
    const int mw = m0 + wave * 16;

    // Staging assignment (coalesced 16B global loads):
    //   A: 128 rows x 16 -> 2 threads/row, 8 floats each (two float4)
    //   B: 16 rows x 64  -> 16 threads/row, 4 floats each (one float4)
    const int ar = tid >> 1;          // 0..127
    const int as = (tid & 1) * 8;     // 0 or 8
    const int br = tid >> 4;          // 0..15 (k within chunk)
    const int bc = (tid & 15) * 4;    // 0..60 (col within tile)

    const float* aptr = X  + (size_t)(m0 + ar) * DD + as;   // advance by +k
    const float* bptr = Wx + (size_t)br * HH + n0 + bc;     // advance by +k*HH

    v8f c0 = {}, c1 = {}, c2 = {}, c3 = {};

    // Register-prefetch chunk 0
    v4f ra0 = *(const v4f*)(aptr);
    v4f ra1 = *(const v4f*)(aptr + 4);
    v4f rb  = *(const v4f*)(bptr);

    for (int k = 0; k < DD; k += KC) {
        // Commit prefetched chunk to LDS
        *(v4f*)(&lA[(size_t)ar * LPAD + as])     = ra0;
        *(v4f*)(&lA[(size_t)ar * LPAD + as + 4]) = ra1;
        lBT[(size_t)(bc + 0) * LPAD + br] = rb.x;
        lBT[(size_t)(bc + 1) * LPAD + br] = rb.y;
        lBT[(size_t)(bc + 2) * LPAD + br] = rb.z;
        lBT[(size_t)(bc + 3) * LPAD + br] = rb.w;
        __syncthreads();

        // Launch global loads for next chunk (in flight during the WMMAs)
        if (k + KC < DD) {
            ra0 = *(const v4f*)(aptr + k + KC);
            ra1 = *(const v4f*)(aptr + k + KC + 4);
            rb  = *(const v4f*)(bptr + (size_t)(k + KC) * HH);
            __builtin_prefetch(aptr + k + 2 * KC, 0, 1);        // warm chunk i+2
            __builtin_prefetch(bptr + (size_t)(k + 2 * KC) * HH, 0, 1);
        }

#pragma unroll
        for (int kk = 0; kk < KC; kk += 4) {
            v2f a = *(const v2f*)(&lA[(size_t)(wave * 16 + lr) * LPAD + kk + 2 * half]);
            WMMA_STEP(c0, 0);
            WMMA_STEP(c1, 16);
            WMMA_STEP(c2, 32);
            WMMA_STEP(c3, 48);
        }
        __syncthreads();  // protect LDS tile before next commit
    }

#define EPILOG_XW(cacc, j)                                                   \
    do {                                                                     \
        const int col = n0 + (j)*16 + lr;                                    \
        const float bb = bias[col];                                          \
        _Pragma("unroll")                                                    \
        for (int e = 0; e < 8; ++e) {                                        \
            XW[(size_t)(mw + e + 8 * half) * HH + col] = cacc[e] + bb;       \
        }                                                                    \
    } while (0)

    EPILOG_XW(c0, 0);
    EPILOG_XW(c1, 1);
    EPILOG_XW(c2, 2);
    EPILOG_XW(c3, 3);
#undef EPILOG_XW
}

// ---------------------------------------------------------------------------
// Kernel 2: one recurrence step.
//   Out_t[n, h] = tanh( sum_k Hprev[n, k] * Wh[k, h] + XWt[n, h] )
// Hprev row stride ld_h (HH for h0, TT*HH when reading out[:, t-1, :]).
// XWt / Out_t row stride TT*HH (views into (N,T,H) at time t).
// Block: 256 threads = 8 waves. Block tile 64(M) x 128(N);
// wave w -> rows (w&3)*16, cols (w>>2)*64 (4 accumulators).
// Grid: (HH/128) = 4 blocks -> 32 waves cover the 64 x 512 step output.
// ---------------------------------------------------------------------------
__global__ __launch_bounds__(256) void rnn_step_kernel(
    const float* __restrict__ Hprev, int ld_h,
    const float* __restrict__ Wh,    // (HH, HH)
    const float* __restrict__ XWt,   // row stride ld_x
    int ld_x,
    float* __restrict__ Outt)        // row stride ld_x
{
    __shared__ float lA[64 * LPAD];   // H tile: [row][k]
    __shared__ float lBT[128 * LPAD]; // Wh tile transposed: [col][k]

    const int tid  = threadIdx.x;
    const int wave = tid >> 5;
    const int lane = tid & 31;
    const int half = lane >> 4;
    const int lr   = lane & 15;

    const int mw = (wave & 3) * 16;                 // rows within M=64
    const int n0 = blockIdx.x * 128;                // block col base
    const int nw = (wave >> 2) * 64;                // wave col base in tile

    // Staging:
    //   A: 64 rows x 16 -> 4 threads/row, one float4 each
    //   B: 16 rows x 128 -> 16 threads/row, 8 floats each (two float4)
    const int ar = tid >> 2;          // 0..63
    const int as = (tid & 3) * 4;     // 0,4,8,12
    const int br = tid >> 4;          // 0..15
    const int bc = (tid & 15) * 8;    // 0..120

    const float* aptr = Hprev + (size_t)ar * ld_h + as;
    const float* bptr = Wh + (size_t)br * HH + n0 + bc;

    v8f c0 = {}, c1 = {}, c2 = {}, c3 = {};

    v4f ra  = *(const v4f*)(aptr);
    v4f rb0 = *(const v4f*)(bptr);
    v4f rb1 = *(const v4f*)(bptr + 4);

    for (int k = 0; k < HH; k += KC) {
        *(v4f*)(&lA[(size_t)ar * LPAD + as]) = ra;
        lBT[(size_t)(bc + 0) * LPAD + br] = rb0.x;
        lBT[(size_t)(bc + 1) * LPAD + br] = rb0.y;
        lBT[(size_t)(bc + 2) * LPAD + br] = rb0.z;
        lBT[(size_t)(bc + 3) * LPAD + br] = rb0.w;
        lBT[(size_t)(bc + 4) * LPAD + br] = rb1.x;
        lBT[(size_t)(bc + 5) * LPAD + br] = rb1.y;
        lBT[(size_t)(bc + 6) * LPAD + br] = rb1.z;
        lBT[(size_t)(bc + 7) * LPAD + br] = rb1.w;
        __syncthreads();

        if (k + KC < HH) {
            ra  = *(const v4f*)(aptr + k + KC);
            rb0 = *(const v4f*)(bptr + (size_t)(k + KC) * HH);
            rb1 = *(const v4f*)(bptr + (size_t)(k + KC) * HH + 4);
        }

#pragma unroll
        for (int kk = 0; kk < KC; kk += 4) {
            v2f a = *(const v2f*)(&lA[(size_t)(mw + lr) * LPAD + kk + 2 * half]);
            WMMA_STEP(c0, nw + 0);
            WMMA_STEP(c1, nw + 16);
            WMMA_STEP(c2, nw + 32);
            WMMA_STEP(c3, nw + 48);
        }
        __syncthreads();
    }

#define EPILOG_STEP(cacc, j)                                                 \
    do {                                                                     \
        const int col = n0 + nw + (j)*16 + lr;                               \
        _Pragma("unroll")                                                    \
        for (int e = 0; e < 8; ++e) {                                        \
            const size_t idx = (size_t)(mw + e + 8 * half) * ld_x + col;     \
            Outt[idx] = tanhf(cacc[e] + XWt[idx]);                           \
        }                                                                    \
    } while (0)

    EPILOG_STEP(c0, 0);
    EPILOG_STEP(c1, 1);
    EPILOG_STEP(c2, 2);
    EPILOG_STEP(c3, 3);
#undef EPILOG_STEP
}

// ---------------------------------------------------------------------------
// Host launcher
// Inputs (setup_inputs order): x (N,T,D), h0 (N,H), Wx (D,H), Wh (H,H), b (H)
// Output: (N, T, H) fp32.  Workspace: xw buffer, N*T*H*4 = 64 MB.
// ---------------------------------------------------------------------------
extern "C" void kernel_launch(void* const* d_in, const int* in_sizes, int n_in,
                              void* d_out, int out_size, void* d_ws, size_t ws_size,
                              hipStream_t stream) {
    const float* x  = (const float*)d_in[0];
    const float* h0 = (const float*)d_in[1];
    const float* Wx = (const float*)d_in[2];
    const float* Wh = (const float*)d_in[3];
    const float* b  = (const float*)d_in[4];
    float* out = (float*)d_out;
    float* xw  = (float*)d_ws;   // (N*T, H) fp32 scratch

    // Phase 1: big parallel GEMM  xw = x @ Wx + b
    {
        dim3 grid(HH / 64, (NB * TT) / 128);
        xw_gemm_kernel<<<grid, 256, 0, stream>>>(x, Wx, b, xw);
    }

    // Phase 2: sequential scan. Step t reads h_{t-1} from out[:, t-1, :]
    // (h0 for t=0) — stream ordering provides the dependency.
    for (int t = 0; t < TT; ++t) {
        const float* hp = (t == 0) ? h0 : (out + (size_t)(t - 1) * HH);
        const int ld_h  = (t == 0) ? HH : TT * HH;
        rnn_step_kernel<<<dim3(HH / 128), 256, 0, stream>>>(
            hp, ld_h, Wh,
            xw + (size_t)t * HH, TT * HH,
            out + (size_t)t * HH);
    }
}